// Binder_MiniGrid_24567212933893
// MI455X (gfx1250) — compile-verified
//
#include <hip/hip_runtime.h>
#include <hip/hip_bf16.h>
#include <math.h>

typedef __attribute__((ext_vector_type(2))) float v2f;
typedef __attribute__((ext_vector_type(8))) float v8f;
typedef __attribute__((ext_vector_type(4))) unsigned int u32x4;
typedef __attribute__((ext_vector_type(4))) int i32x4;
typedef __attribute__((ext_vector_type(8))) int i32x8;

#define EPSV  1e-5f
#define NB    256
#define CIN   32
#define HIN   48
#define WIN   48
#define HO    41
#define WO    41
#define LPOS  (HO*WO)        /* 1681 */
#define EDIM  256
#define KTOT  2048           /* CIN*8*8 */
#define MAXP  8
#define MTILE 32
#define NBLK  53             /* ceil(1681/32) */

/* workspace layout (float offsets) */
#define OFF_WKG    0
#define OFF_CONSTS 256
#define OFF_SCORES 512
#define OFF_CNT    (OFF_SCORES + NB*LPOS)
#define OFF_POS    (OFF_CNT + NB)
#define OFF_WGT    (OFF_POS + NB*MAXP)

/* ---------------- kernel 0: fold q=bq into the k-projection ---------------- */
__global__ __launch_bounds__(256)
void precompute_kernel(const float* __restrict__ bq, const float* __restrict__ wk,
                       const float* __restrict__ bk, const float* __restrict__ g1,
                       const float* __restrict__ b1, float* __restrict__ ws) {
    __shared__ float buf[256];
    const int e = threadIdx.x;
    float kq = 0.f;
    for (int f = 0; f < EDIM; ++f) kq += bq[f] * wk[f*EDIM + e];
    const float wkg = kq * g1[e];
    ws[OFF_WKG + e] = wkg;
    buf[e] = wkg; __syncthreads();
    for (int s = 128; s > 0; s >>= 1) { if (e < s) buf[e] += buf[e+s]; __syncthreads(); }
    const float A = buf[0]; __syncthreads();
    buf[e] = b1[e]*kq + bq[e]*bk[e]; __syncthreads();
    for (int s = 128; s > 0; s >>= 1) { if (e < s) buf[e] += buf[e+s]; __syncthreads(); }
    if (e == 0) { ws[OFF_CONSTS+0] = A; ws[OFF_CONSTS+1] = buf[0]; }
}

/* ------------- kernel 1: implicit-GEMM conv (fp32 WMMA) -> scores ---------- */
__global__ __launch_bounds__(256)
void conv_score_kernel(const float* __restrict__ Xc, const float* __restrict__ Xt,
                       const float* __restrict__ W,  const float* __restrict__ cb,
                       float* __restrict__ ws) {
    __shared__ float A_lds[MTILE * 68];   /* 32 x 64, pad->68 for conflict-free b64 reads */
    __shared__ float B_lds[EDIM * 68];    /* 256 x 64, pad->68 (matches TDM pad 64+4 dw) */
    __shared__ float r_dot[MTILE], r_sum[MTILE], r_sq[MTILE];

    const int b    = blockIdx.y;
    const int tile = blockIdx.x;
    const int t    = threadIdx.x;
    const int wave = t >> 5;
    const int lane = t & 31;
    const int half = lane >> 4;
    const int lr   = lane & 15;
    const int n0   = wave * 32;

    const float* X = (b < 128) ? (Xc + (size_t)b * CIN*HIN*WIN)
                               : (Xt + (size_t)(b-128) * CIN*HIN*WIN);

    /* A staging assignment: thread -> (row m, ky) */
    const int am  = t & 31;
    const int aky = t >> 5;
    int l = tile*MTILE + am; if (l > LPOS-1) l = LPOS-1;   /* clamp tail (guarded at write) */
    const int py = l / WO, px = l % WO;

    /* B staging assignment (fallback path): thread -> (e-row group, 4-wide k slice) */
    const int be0 = t >> 4;
    const int bk4 = (t & 15) * 4;

    v8f acc00 = {}, acc01 = {}, acc10 = {}, acc11 = {};

    for (int c = 0; c < CIN; ++c) {
        __syncthreads();
        /* stage A tile: 32 positions x 64 (ky,kx) for channel c */
        {
            const float* xr = X + ((size_t)c*HIN + (py + aky))*WIN + px;
            float4 v0 = make_float4(xr[0], xr[1], xr[2], xr[3]);
            float4 v1 = make_float4(xr[4], xr[5], xr[6], xr[7]);
            float4* ad = (float4*)&A_lds[am*68 + aky*8];
            ad[0] = v0; ad[1] = v1;
        }
        /* stage B tile: 256 e-rows x 64 k for channel c */
#if __has_builtin(__builtin_amdgcn_tensor_load_to_lds)
        if (t < 32) {   /* wave 0 issues one TDM descriptor for the whole tile */
            const unsigned long long ga =
                (unsigned long long)(uintptr_t)(W + (size_t)c * 64);
            const unsigned int la = (unsigned int)(uintptr_t)(&B_lds[0]); /* LDS byte offset */
            u32x4 g0;
            g0[0] = 1u;                                   /* count=1, gather off      */
            g0[1] = la;                                   /* lds_addr                 */
            g0[2] = (unsigned int)ga;                     /* global_addr[31:0]        */
            g0[3] = (unsigned int)(ga >> 32) | (2u << 30);/* global_addr[56:32]|type=2*/
            i32x8 g1;
            g1[0] = (int)((2u << 16)    /* data_size = 4B            */
                        | (1u << 20)    /* pad_enable                */
                        | (5u << 22)    /* pad_interval: every 64 dw */
                        | (3u << 25));  /* pad_amount: 4 dwords      */
            g1[1] = (int)(2048u << 16); /* tensor_dim0 = 2048 (lo16 @bits63:48) */
            g1[2] = (int)(256u  << 16); /* tensor_dim0 hi=0 | tensor_dim1 = 256 */
            g1[3] = (int)(64u   << 16); /* tensor_dim1 hi=0 | tile_dim0 = 64    */
            g1[4] = 256;                /* tile_dim1 = 256, tile_dim2 = 0       */
            g1[5] = 2048;               /* tensor_dim0_stride lo32              */
            g1[6] = 0;                  /* stride hi | dim1_stride lo           */
            g1[7] = 0;
            i32x4 g2 = {}; i32x4 g3 = {};
#if __clang_major__ >= 23
            i32x8 g4 = {};
            __builtin_amdgcn_tensor_load_to_lds(g0, g1, g2, g3, g4, 0);
#else
            __builtin_amdgcn_tensor_load_to_lds(g0, g1, g2, g3, 0);
#endif
#if __has_builtin(__builtin_amdgcn_s_wait_tensorcnt)
            __builtin_amdgcn_s_wait_tensorcnt(0);
#else
            asm volatile("s_wait_tensorcnt 0x0" ::: "memory");
#endif
        }
#else
        for (int p = 0; p < 16; ++p) {
            const int e = p*16 + be0;
            const float4 wv4 = *(const float4*)&W[((size_t)e*CIN + c)*64 + bk4];
            *(float4*)&B_lds[e*68 + bk4] = wv4;
        }
#endif
        __syncthreads();

        for (int j = 0; j < 64; j += 4) {
            const int k0 = j + half*2;     /* K=0,1 lanes 0-15 ; K=2,3 lanes 16-31 */
            v2f a0f, a1f, b0f, b1f;
            a0f.x = A_lds[lr*68 + k0];        a0f.y = A_lds[lr*68 + k0 + 1];
            a1f.x = A_lds[(lr+16)*68 + k0];   a1f.y = A_lds[(lr+16)*68 + k0 + 1];
            b0f.x = B_lds[(n0+lr)*68 + k0];   b0f.y = B_lds[(n0+lr)*68 + k0 + 1];
            b1f.x = B_lds[(n0+16+lr)*68 + k0];b1f.y = B_lds[(n0+16+lr)*68 + k0 + 1];
            acc00 = __builtin_amdgcn_wmma_f32_16x16x4_f32(false, a0f, false, b0f, (short)0, acc00, false, false);
            acc01 = __builtin_amdgcn_wmma_f32_16x16x4_f32(false, a0f, false, b1f, (short)0, acc01, false, false);
            acc10 = __builtin_amdgcn_wmma_f32_16x16x4_f32(false, a1f, false, b0f, (short)0, acc10, false, false);
            acc11 = __builtin_amdgcn_wmma_f32_16x16x4_f32(false, a1f, false, b1f, (short)0, acc11, false, false);
        }
    }

    __syncthreads();
    if (t < MTILE) { r_dot[t] = 0.f; r_sum[t] = 0.f; r_sq[t] = 0.f; }
    __syncthreads();

    /* epilogue: fold conv bias, reduce over E for LN stats + score dot */
    const float wk0 = ws[OFF_WKG + n0 + lr];
    const float wk1 = ws[OFF_WKG + n0 + 16 + lr];
    const float cb0 = cb[n0 + lr];
    const float cb1 = cb[n0 + 16 + lr];
    for (int r = 0; r < 8; ++r) {
        {
            const float v0 = acc00[r] + cb0, v1 = acc01[r] + cb1;
            const int m = r + 8*half;
            atomicAdd(&r_sum[m], v0 + v1);
            atomicAdd(&r_dot[m], v0*wk0 + v1*wk1);
            atomicAdd(&r_sq[m],  v0*v0 + v1*v1);
        }
        {
            const float v0 = acc10[r] + cb0, v1 = acc11[r] + cb1;
            const int m = 16 + r + 8*half;
            atomicAdd(&r_sum[m], v0 + v1);
            atomicAdd(&r_dot[m], v0*wk0 + v1*wk1);
            atomicAdd(&r_sq[m],  v0*v0 + v1*v1);
        }
    }
    __syncthreads();

    if (t < MTILE) {
        const int lo = tile*MTILE + t;
        if (lo < LPOS) {
            const float Ac = ws[OFF_CONSTS+0];
            const float Bc = ws[OFF_CONSTS+1];
            const float mu  = r_sum[t] * (1.0f/EDIM);
            const float var = r_sq[t] * (1.0f/EDIM) - mu*mu;
            const float rs  = rsqrtf(var + EPSV);
            ws[OFF_SCORES + (size_t)b*LPOS + lo] =
                (rs * (r_dot[t] - mu*Ac) + Bc) * (1.0f/16.0f);
        }
    }
}

/* ---------------- kernel 2: per-batch top-4 + masked softmax --------------- */
__global__ __launch_bounds__(256)
void topk_kernel(float* __restrict__ ws) {
    __shared__ float s_orig[LPOS];
    __shared__ float s_work[LPOS];
    __shared__ float rv[256];
    __shared__ int   ri[256];
    __shared__ int   cnt_sh;
    const int b = blockIdx.x;
    const int t = threadIdx.x;
    const float* sc = ws + OFF_SCORES + (size_t)b*LPOS;
    for (int l = t; l < LPOS; l += 256) { const float v = sc[l]; s_orig[l] = v; s_work[l] = v; }
    if (t == 0) cnt_sh = 0;
    __syncthreads();

    float top[4];
    for (int kk = 0; kk < 4; ++kk) {
        float lv = -INFINITY; int li = -1;
        for (int l = t; l < LPOS; l += 256) { const float v = s_work[l]; if (v > lv) { lv = v; li = l; } }
        rv[t] = lv; ri[t] = li; __syncthreads();
        for (int s = 128; s > 0; s >>= 1) {
            if (t < s && rv[t+s] > rv[t]) { rv[t] = rv[t+s]; ri[t] = ri[t+s]; }
            __syncthreads();
        }
        top[kk] = rv[0];
        if (t == 0 && ri[0] >= 0) s_work[ri[0]] = -INFINITY;
        __syncthreads();
    }
    const float vmax = top[0], kth = top[3];

    float lz = 0.f;
    for (int l = t; l < LPOS; l += 256) { const float v = s_orig[l]; if (v >= kth) lz += expf(v - vmax); }
    rv[t] = lz; __syncthreads();
    for (int s = 128; s > 0; s >>= 1) { if (t < s) rv[t] += rv[t+s]; __syncthreads(); }
    const float Z = rv[0];
    __syncthreads();

    int*   cnt = (int*)(ws + OFF_CNT);
    int*   pos = (int*)(ws + OFF_POS);
    float* wgt = ws + OFF_WGT;
    for (int l = t; l < LPOS; l += 256) {
        const float v = s_orig[l];
        if (v >= kth) {
            const int slot = atomicAdd(&cnt_sh, 1);
            if (slot < MAXP) { pos[b*MAXP + slot] = l; wgt[b*MAXP + slot] = expf(v - vmax) / Z; }
        }
    }
    __syncthreads();
    if (t == 0) cnt[b] = min(cnt_sh, MAXP);
}

/* ---- kernel 3: recompute conv at winners, LN, xWvT+bv, LN2, ReLU, pack ---- */
__global__ __launch_bounds__(256)
void gather_out_kernel(const float* __restrict__ Xc, const float* __restrict__ Xt,
                       const float* __restrict__ W,  const float* __restrict__ cb,
                       const float* __restrict__ g1, const float* __restrict__ b1,
                       const float* __restrict__ wv, const float* __restrict__ bv,
                       const float* __restrict__ g2, const float* __restrict__ b2,
                       const float* __restrict__ ws, float* __restrict__ out) {
    __shared__ float patch[KTOT];
    __shared__ float red[256];
    __shared__ float g_lds[EDIM];
    const int b = blockIdx.x;
    const int t = threadIdx.x;
    const float* X = (b < 128) ? (Xc + (size_t)b * CIN*HIN*WIN)
                               : (Xt + (size_t)(b-128) * CIN*HIN*WIN);
    const int*   cnt = (const int*)(ws + OFF_CNT);
    const int*   pos = (const int*)(ws + OFF_POS);
    const float* wgt = ws + OFF_WGT;
    const int np = cnt[b];

    float g = 0.f;
    for (int p = 0; p < np; ++p) {
        const int   l = pos[b*MAXP + p];
        const float w = wgt[b*MAXP + p];
        const int py = l / WO, px = l % WO;
        {   /* stage 32x8x8 input patch */
            const int pc = t >> 3, pky = t & 7;
            const float* xr = X + ((size_t)pc*HIN + (py + pky))*WIN + px;
            float* pd = &patch[pc*64 + pky*8];
            for (int u = 0; u < 8; ++u) pd[u] = xr[u];
        }
        __syncthreads();
        const float* wrow = W + (size_t)t * KTOT;
        float f = cb[t];
        for (int k = 0; k < KTOT; ++k) f += patch[k] * wrow[k];
        red[t] = f; __syncthreads();
        for (int s = 128; s > 0; s >>= 1) { if (t < s) red[t] += red[t+s]; __syncthreads(); }
        const float mu = red[0] * (1.0f/EDIM); __syncthreads();
        const float d = f - mu;
        red[t] = d*d; __syncthreads();
        for (int s = 128; s > 0; s >>= 1) { if (t < s) red[t] += red[t+s]; __syncthreads(); }
        const float var = red[0] * (1.0f/EDIM); __syncthreads();
        g += w * (d * rsqrtf(var + EPSV) * g1[t] + b1[t]);
        __syncthreads();   /* protect patch before next overwrite */
    }
    g_lds[t] = g; __syncthreads();

    const float* wvr = wv + (size_t)t * EDIM;
    float o = bv[t];
    for (int e = 0; e < EDIM; ++e) o += g_lds[e] * wvr[e];

    red[t] = o; __syncthreads();
    for (int s = 128; s > 0; s >>= 1) { if (t < s) red[t] += red[t+s]; __syncthreads(); }
    const float mu = red[0] * (1.0f/EDIM); __syncthreads();
    const float d = o - mu;
    red[t] = d*d; __syncthreads();
    for (int s = 128; s > 0; s >>= 1) { if (t < s) red[t] += red[t+s]; __syncthreads(); }
    const float var = red[0] * (1.0f/EDIM); __syncthreads();
    float val = d * rsqrtf(var + EPSV) * g2[t] + b2[t];
    val = fmaxf(val, 0.f);
    if (b < 128) out[(size_t)b*512 + t]             = val;
    else         out[(size_t)(b-128)*512 + 256 + t] = val;
}

extern "C" void kernel_launch(void* const* d_in, const int* in_sizes, int n_in,
                              void* d_out, int out_size, void* d_ws, size_t ws_size,
                              hipStream_t stream) {
    const float* Xc = (const float*)d_in[0];
    const float* Xt = (const float*)d_in[1];
    const float* W  = (const float*)d_in[2];
    const float* cb = (const float*)d_in[3];
    const float* g1 = (const float*)d_in[4];
    const float* b1 = (const float*)d_in[5];
    /* d_in[6] = wq : unused (q == bq) */
    const float* bq = (const float*)d_in[7];
    const float* wk = (const float*)d_in[8];
    const float* bk = (const float*)d_in[9];
    const float* wv = (const float*)d_in[10];
    const float* bv = (const float*)d_in[11];
    const float* g2 = (const float*)d_in[12];
    const float* b2 = (const float*)d_in[13];
    float* ws  = (float*)d_ws;
    float* out = (float*)d_out;

    precompute_kernel<<<1, 256, 0, stream>>>(bq, wk, bk, g1, b1, ws);
    conv_score_kernel<<<dim3(NBLK, NB), 256, 0, stream>>>(Xc, Xt, W, cb, ws);
    topk_kernel<<<NB, 256, 0, stream>>>(ws);
    gather_out_kernel<<<NB, 256, 0, stream>>>(Xc, Xt, W, cb, g1, b1, wv, bv, g2, b2, ws, out);
}